// SLICLayer_70162585747543
// MI455X (gfx1250) — compile-verified
//
#include <hip/hip_runtime.h>

typedef __attribute__((ext_vector_type(2))) float v2f;
typedef __attribute__((ext_vector_type(8))) float v8f;

#define NSEG   100
#define KT     7                 // ceil(100/16) center tiles
#define PPIX   50176             // 224*224 (multiple of 16)
#define TOTPIX (16 * PPIX)       // 802816 pixels
#define NGROUP (TOTPIX / 16)     // 50176 groups of 16 pixels
#define ACC_OFF_BYTES 4096       // f64 accumulators live here inside d_ws

// ---------------------------------------------------------------------------
// init: centers -> ws[0..299], zero the 100x4 double accumulators
// ---------------------------------------------------------------------------
__global__ __launch_bounds__(512)
void slic_init(const float* __restrict__ cin, float* __restrict__ ws)
{
    int i = threadIdx.x;
    if (i < NSEG * 3) ws[i] = cin[i];
    double* gacc = (double*)((char*)ws + ACC_OFF_BYTES);
    if (i < NSEG * 4) gacc[i] = 0.0;
}

// ---------------------------------------------------------------------------
// assign: per wave, WMMA f32 16x16x4 computes 16 pixels x 16 centers dots.
// score = |c|^2 - 2*dot  (|p|^2 constant per pixel -> dropped for argmin).
// Labels written to out; feature sums/counts accumulated in f64.
// ---------------------------------------------------------------------------
__global__ __launch_bounds__(256)
void slic_assign(const float* __restrict__ img, float* __restrict__ ws,
                 float* __restrict__ out)
{
    __shared__ v2f    bfrag[KT][32];   // B fragments per tile, per lane
    __shared__ float  csqt[KT][16];    // |center|^2 per tile column
    __shared__ double acc[NSEG * 4];   // block-local f64 sums+count

    const float* centers = ws;
    double* gacc = (double*)((char*)ws + ACC_OFF_BYTES);

    const int tid = threadIdx.x;

    // Build B fragments: B[k][n] = centers[n][k], K padded to 4, N to 112.
    // Assumed 32-bit 4x16 B layout mirroring the A-matrix table:
    //   lanes 0-15:  {K=0,K=1} for column n=lane
    //   lanes 16-31: {K=2,K=3} for column n=lane-16
    if (tid < KT * 32) {
        int t  = tid >> 5, ln = tid & 31;
        int cc = ln & 15, hf = ln >> 4;
        int n  = cc + 16 * t;
        v2f b;
        if (n < NSEG) {
            if (hf == 0) { b.x = centers[n*3+0]; b.y = centers[n*3+1]; }
            else         { b.x = centers[n*3+2]; b.y = 0.0f; }
        } else {
            b.x = 0.0f; b.y = 0.0f;
        }
        bfrag[t][ln] = b;
        if (hf == 0) {
            float cs = 3.0e38f;                  // padded centers never win
            if (n < NSEG) {
                float c0 = centers[n*3+0], c1 = centers[n*3+1], c2 = centers[n*3+2];
                cs = c0*c0 + c1*c1 + c2*c2;
            }
            csqt[t][cc] = cs;
        }
    }
    for (int i = tid; i < NSEG * 4; i += 256) acc[i] = 0.0;
    __syncthreads();

    const int lane = tid & 31;
    const int wave = tid >> 5;
    const int col  = lane & 15;
    const int hf   = lane >> 4;
    const int gw0  = blockIdx.x * 8 + wave;
    const int nw   = gridDim.x * 8;

    for (int g = gw0; g < NGROUP; g += nw) {
        const int base = g << 4;                 // first pixel of group
        const int bi   = base / PPIX;            // image index (group-uniform)
        const int p0   = base - bi * PPIX;       // pixel offset within image
        const float* ib = img + (size_t)bi * 3 * PPIX;

        // A fragment (32-bit 16x4 layout):
        //   lanes 0-15:  {ch0, ch1} of pixel m=lane
        //   lanes 16-31: {ch2, 0}   of pixel m=lane-16
        v2f a;
        {
            int p = p0 + col;
            if (hf == 0) { a.x = ib[p];            a.y = ib[PPIX + p]; }
            else         { a.x = ib[2*PPIX + p];   a.y = 0.0f;         }
        }

        float bestv[8];
        int   bestn[8];
#pragma unroll
        for (int r = 0; r < 8; ++r) { bestv[r] = 3.0e38f; bestn[r] = 0; }

#pragma unroll
        for (int t = 0; t < KT; ++t) {
            v2f b  = bfrag[t][lane];
            float cs = csqt[t][col];
            v8f cz = {};
            v8f d = __builtin_amdgcn_wmma_f32_16x16x4_f32(
                        false, a, false, b, (short)0, cz, false, false);
            int n = col + 16 * t;
#pragma unroll
            for (int r = 0; r < 8; ++r) {
                float s = __builtin_fmaf(-2.0f, d[r], cs);
                if (s < bestv[r]) { bestv[r] = s; bestn[r] = n; }
            }
        }

        // Butterfly argmin within each 16-lane half (xor masks < 16 keep
        // lanes inside their half). Tie-break: smaller center index, to
        // match jnp.argmin's first-minimum semantics.
#pragma unroll
        for (int r = 0; r < 8; ++r) {
            float v = bestv[r]; int n = bestn[r];
#pragma unroll
            for (int off = 1; off < 16; off <<= 1) {
                float vv = __shfl_xor(v, off, 32);
                int   nn = __shfl_xor(n, off, 32);
                if (vv < v || (vv == v && nn < n)) { v = vv; n = nn; }
            }
            bestv[r] = v; bestn[r] = n;
        }

        // D layout: VGPR r holds M=r (lanes 0-15) and M=8+r (lanes 16-31).
        // Lanes 0 and 16 commit 8 pixels each.
        if (col == 0) {
            const int prow = hf * 8;
#pragma unroll
            for (int r = 0; r < 8; ++r) {
                int k  = bestn[r];
                int pp = p0 + prow + r;
                int q  = base + prow + r;
                out[q] = (float)k;
                float f0 = ib[pp], f1 = ib[PPIX + pp], f2 = ib[2*PPIX + pp];
                atomicAdd(&acc[k*4+0], (double)f0);
                atomicAdd(&acc[k*4+1], (double)f1);
                atomicAdd(&acc[k*4+2], (double)f2);
                atomicAdd(&acc[k*4+3], 1.0);
            }
        }
    }

    __syncthreads();
    for (int i = tid; i < NSEG * 4; i += 256) {
        double v = acc[i];
        if (v != 0.0) atomicAdd(&gacc[i], v);
    }
}

// ---------------------------------------------------------------------------
// update: centers = where(cnt>0, sum/max(cnt,1), 0); zero accumulators
// ---------------------------------------------------------------------------
__global__ __launch_bounds__(128)
void slic_update(float* __restrict__ ws)
{
    double* gacc = (double*)((char*)ws + ACC_OFF_BYTES);
    int k = threadIdx.x;
    if (k < NSEG) {
        double s0 = gacc[k*4+0], s1 = gacc[k*4+1], s2 = gacc[k*4+2];
        double c  = gacc[k*4+3];
        double inv = 1.0 / (c > 1.0 ? c : 1.0);
        bool nz = (c > 0.0);
        ws[k*3+0] = nz ? (float)(s0 * inv) : 0.0f;
        ws[k*3+1] = nz ? (float)(s1 * inv) : 0.0f;
        ws[k*3+2] = nz ? (float)(s2 * inv) : 0.0f;
        gacc[k*4+0] = 0.0; gacc[k*4+1] = 0.0;
        gacc[k*4+2] = 0.0; gacc[k*4+3] = 0.0;
    }
}

// ---------------------------------------------------------------------------
extern "C" void kernel_launch(void* const* d_in, const int* in_sizes, int n_in,
                              void* d_out, int out_size, void* d_ws, size_t ws_size,
                              hipStream_t stream)
{
    (void)in_sizes; (void)n_in; (void)out_size; (void)ws_size;
    const float* img = (const float*)d_in[0];
    const float* cin = (const float*)d_in[1];
    float* ws  = (float*)d_ws;
    float* out = (float*)d_out;

    slic_init<<<1, 512, 0, stream>>>(cin, ws);
    for (int it = 0; it < 10; ++it) {
        slic_assign<<<784, 256, 0, stream>>>(img, ws, out);
        slic_update<<<1, 128, 0, stream>>>(ws);
    }
}